// incremental_HMM_7473243095171
// MI455X (gfx1250) — compile-verified
//
#include <hip/hip_runtime.h>

// CDNA5 / gfx1250 — wave32, WMMA f32 16x16x4 + Tensor Data Mover staging.
typedef float v2f __attribute__((ext_vector_type(2)));
typedef float v8f __attribute__((ext_vector_type(8)));
typedef unsigned int v4u __attribute__((ext_vector_type(4)));
typedef unsigned int v8u __attribute__((ext_vector_type(8)));

#define R_ 32
#define N_ 8192
#define LEN_ 2048
#define WAVES 8       // waves per block in main kernel (256 threads)
#define TCHUNK 64     // time steps per block  -> 64*128*4B = 32 KB LDS tile
#define TCHUNKS (LEN_ / TCHUNK)

// ---------------------------------------------------------------------------
// Kernel 1: one block, 1024 threads (32 waves).
//  - squares transition 6x in LDS -> T^64
//  - wave w walks seed h(64w) = init_w @ (T^64)^w  (shfl broadcasts)
//  - emits A-operands PRE-PACKED in WMMA A-layout (base-2 log domain):
//      Apk[t][tile][lane<16]  = ( C2[t][16*tile+lane], B2[t][16*tile+lane] )
//      Apk[t][tile][lane>=16] = ( E2[16*tile+lane-16], 0 )
//    with C2 = log2 h - log2 Σh - log2 σ - ½log2(2π) - log2e·(tμ)²/(2σ²),
//         B2 = log2e·tμ/σ²,  E2 = -log2e/(2σ²).
// ---------------------------------------------------------------------------
__global__ __launch_bounds__(1024) void hmm_setup_kernel(
    const float* __restrict__ transition, const float* __restrict__ init_w,
    const float* __restrict__ sig, const float* __restrict__ mu_rates,
    float* __restrict__ Apk /* [LEN][2][32] float2 */) {
  __shared__ float T[R_][R_];
  __shared__ float P[2][R_][R_];
  const int tid = threadIdx.x;
  const int r = tid & 31;
  const int row = tid >> 5;  // 0..31 == wave index (wave32)

  T[row][r] = transition[row * R_ + r];
  __syncthreads();

  // P[0] = T^2
  {
    float a = 0.f;
    #pragma unroll 4
    for (int k = 0; k < R_; ++k) a += T[row][k] * T[k][r];
    P[0][row][r] = a;
  }
  __syncthreads();
  // 5 more squarings -> P[1] = T^64
  for (int s = 1; s < 6; ++s) {
    const float(*src)[R_] = P[(s - 1) & 1];
    float a = 0.f;
    #pragma unroll 4
    for (int k = 0; k < R_; ++k) a += src[row][k] * src[k][r];
    P[s & 1][row][r] = a;
    __syncthreads();
  }

  // Seed h at t = 64*row (each wave redundantly; no cross-wave sync needed)
  const int w = row;
  float h = init_w[r];
  for (int c = 0; c < w; ++c) {
    float hn = 0.f;
    #pragma unroll 4
    for (int k = 0; k < R_; ++k) hn += __shfl(h, k, 32) * P[1][k][r];
    h = hn;
  }

  const float LOG2E = 1.44269504088896340736f;
  const float HALF_LOG2_2PI = 1.32574806473616002f;  // 0.5*log2(2*pi)
  const float sr = sig[r];
  const float log2sig = __builtin_amdgcn_logf(sr);
  const float inv_s2 = 1.0f / (sr * sr);
  const float mur = mu_rates[r];
  const float ecr = -0.5f * LOG2E * inv_s2;

  float2* Apk2 = reinterpret_cast<float2*>(Apk);
  const int tile = r >> 4;
  const int slot = r & 15;
  for (int s = 0; s < 64; ++s) {
    const int t = w * 64 + s;
    float sh = h;
    #pragma unroll
    for (int off = 16; off >= 1; off >>= 1) sh += __shfl_xor(sh, off, 32);
    const float mu_t = (float)t * mur;
    float2 cm, ce;
    cm.x = __builtin_amdgcn_logf(h) - __builtin_amdgcn_logf(sh) - log2sig -
           HALF_LOG2_2PI - 0.5f * LOG2E * inv_s2 * mu_t * mu_t;
    cm.y = LOG2E * inv_s2 * mu_t;
    ce.x = ecr;
    ce.y = 0.f;
    Apk2[t * 64 + tile * 32 + slot] = cm;       // A VGPR pair, lanes 0..15
    Apk2[t * 64 + tile * 32 + 16 + slot] = ce;  // A VGPR pair, lanes 16..31
    // h <- h @ T (faithful per-step recurrence inside the chunk)
    float hn = 0.f;
    #pragma unroll 4
    for (int k = 0; k < R_; ++k) hn += __shfl(h, k, 32) * T[k][r];
    h = hn;
  }
}

// ---------------------------------------------------------------------------
// Kernel 2: main. Block stages its 64-t coefficient chunk into LDS with ONE
// tensor_load_to_lds (TDM, D# groups 0/1; 2D tile 128 x 64 f32, contiguous),
// waits on TENSORcnt, barriers, then each wave owns a 16-row n-tile:
// per t -> 2x V_WMMA_F32_16X16X4_F32 (M=r, N=n, K=[1,x,x2]) with A operands
// read straight from LDS in register layout, then SUM_r exp2 in-lane (+1
// shfl_xor(16)) and accumulate log2.
// ---------------------------------------------------------------------------
__global__ __launch_bounds__(32 * WAVES) void hmm_main_kernel(
    const float* __restrict__ X, const float* __restrict__ Apk,
    float* __restrict__ partials) {
  __shared__ float ldsA[TCHUNK * 128];  // 32 KB, written by TDM
  __shared__ float wsum[WAVES];

  const int lane = threadIdx.x & 31;
  const int wv = threadIdx.x >> 5;
  const int m = lane & 15;
  const bool hi = lane >= 16;
  const int n0 = (blockIdx.x * WAVES + wv) * 16;
  const float* xrow = X + (size_t)(n0 + m) * LEN_;
  const int tbeg = blockIdx.y * TCHUNK;

  // ---- TDM: stage Apk[tbeg .. tbeg+TCHUNK) (contiguous 32 KB) into LDS ----
  if (threadIdx.x == 0) {
    const unsigned long long gaddr =
        (unsigned long long)(uintptr_t)(Apk + (size_t)tbeg * 128);
    v4u g0;
    g0.x = 1u;                                    // count=1 (valid descriptor)
    g0.y = (unsigned)(uintptr_t)&ldsA[0];         // lds_addr (byte offset)
    g0.z = (unsigned)(gaddr & 0xFFFFFFFFull);     // global_addr[31:0]
    g0.w = (unsigned)((gaddr >> 32) & 0x1FFFFFFull) | (2u << 30);  // type=2
    v8u g1;
    g1.s0 = 2u << 16;             // data_size = 4B
    g1.s1 = 128u << 16;           // tensor_dim0 = 128 (elements per t-row)
    g1.s2 = (unsigned)TCHUNK << 16;  // tensor_dim1 = 64
    g1.s3 = 128u << 16;           // tile_dim0 = 128
    g1.s4 = (unsigned)TCHUNK;     // tile_dim1 = 64, tile_dim2 = 0
    g1.s5 = 128u;                 // tensor_dim0_stride = 128
    g1.s6 = (unsigned)(TCHUNK * 128) << 16;  // tensor_dim1_stride lo16
    g1.s7 = 0u;
    asm volatile("tensor_load_to_lds %0, %1" : : "s"(g0), "s"(g1) : "memory");
    __builtin_amdgcn_s_wait_tensorcnt(0);
  }
  __syncthreads();

  const float2* ldsA2 = reinterpret_cast<const float2*>(ldsA);
  const v8f czero = {};
  float accLog = 0.f;

  for (int j0 = 0; j0 < TCHUNK; j0 += 4) {
    const float4 xv = *reinterpret_cast<const float4*>(xrow + tbeg + j0);
    __builtin_prefetch(xrow + tbeg + j0 + 64, 0, 1);  // global_prefetch_b8
    #pragma unroll
    for (int j = 0; j < 4; ++j) {
      const int tt = j0 + j;
      const float x = (&xv.x)[j];
      const float2 a0 = ldsA2[tt * 64 + lane];       // tile r0..15  (ds_load_b64)
      const float2 a1 = ldsA2[tt * 64 + 32 + lane];  // tile r16..31
      v2f A0, A1, B;
      A0[0] = a0.x;  A0[1] = a0.y;                   // already in A layout
      A1[0] = a1.x;  A1[1] = a1.y;
      B[0] = hi ? x * x : 1.0f;                      // B rows [1, x, x2, 0]
      B[1] = hi ? 0.f : x;
      v8f d0 = __builtin_amdgcn_wmma_f32_16x16x4_f32(false, A0, false, B,
                                                     (short)0, czero, false, false);
      v8f d1 = __builtin_amdgcn_wmma_f32_16x16x4_f32(false, A1, false, B,
                                                     (short)0, czero, false, false);
      float s = 0.f;
      #pragma unroll
      for (int q = 0; q < 8; ++q)
        s += __builtin_amdgcn_exp2f(d0[q]) + __builtin_amdgcn_exp2f(d1[q]);
      s += __shfl_xor(s, 16, 32);           // combine r-halves for this n
      accLog += __builtin_amdgcn_logf(s);   // log2; rescaled by ln2 at the end
    }
  }

  // wave reduce (each n counted twice -> halved in final scale)
  #pragma unroll
  for (int off = 16; off >= 1; off >>= 1) accLog += __shfl_xor(accLog, off, 32);
  if (lane == 0) wsum[wv] = accLog;
  __syncthreads();
  if (threadIdx.x == 0) {
    float tot = 0.f;
    #pragma unroll
    for (int i = 0; i < WAVES; ++i) tot += wsum[i];
    partials[blockIdx.y * gridDim.x + blockIdx.x] = tot;
  }
}

// ---------------------------------------------------------------------------
// Kernel 3: deterministic fixed-order double-precision reduction.
// ---------------------------------------------------------------------------
__global__ __launch_bounds__(64) void hmm_reduce_kernel(
    const float* __restrict__ partials, int np, float* __restrict__ out,
    float scale) {
  __shared__ double sbuf[64];
  double acc = 0.0;
  for (int i = threadIdx.x; i < np; i += 64) acc += (double)partials[i];
  sbuf[threadIdx.x] = acc;
  __syncthreads();
  if (threadIdx.x == 0) {
    double tot = 0.0;
    for (int i = 0; i < 64; ++i) tot += sbuf[i];
    *out = (float)(tot * (double)scale);
  }
}

extern "C" void kernel_launch(void* const* d_in, const int* in_sizes, int n_in,
                              void* d_out, int out_size, void* d_ws, size_t ws_size,
                              hipStream_t stream) {
  const float* X          = (const float*)d_in[0];  // [N,1,L]
  const float* transition = (const float*)d_in[1];  // [R,R]
  const float* init_w     = (const float*)d_in[2];  // [1,R]
  const float* sig        = (const float*)d_in[3];  // [1,R]
  const float* mu_rates   = (const float*)d_in[4];  // [1,R]

  // workspace (floats): Apk[L][2][32][2] (1 MB) | partials[2048]
  float* ws = (float*)d_ws;
  float* Apk = ws;
  float* partials = ws + (size_t)LEN_ * 128;

  hmm_setup_kernel<<<dim3(1), dim3(1024), 0, stream>>>(transition, init_w, sig,
                                                       mu_rates, Apk);

  dim3 grid(N_ / (16 * WAVES), TCHUNKS);  // 64 x 32 blocks, 16K waves
  hmm_main_kernel<<<grid, dim3(32 * WAVES), 0, stream>>>(X, Apk, partials);

  const int np = (N_ / (16 * WAVES)) * TCHUNKS;
  const float scale = 0.69314718055994530942f * 0.5f / (float)N_;  // ln2/(2N)
  hmm_reduce_kernel<<<dim3(1), dim3(64), 0, stream>>>(partials, np,
                                                      (float*)d_out, scale);
}